// CBOW_49701361549346
// MI455X (gfx1250) — compile-verified
//
#include <hip/hip_runtime.h>
#include <math.h>

// ---------------------------------------------------------------------------
// CBOW forward on MI455X (gfx1250, wave32)
//   N=4 (2N=8 context), D=64, V=50000, B=256
//
// Pipeline:
//   k_extract : stream one-hot x (409.6MB, 4-way unrolled NT b128 loads),
//               write 2048 (id,val) pairs           -> HBM roofline ~17.6us
//   k_hidden  : h[b,d] = b1[d] + (1/8) * sum_i val*w1[d,id]  (tiny gather)
//   k_stats   : WMMA f32 16x16x4 GEMM tiles (h x w2^T + b2); double-buffered
//               b128 B-fragment loads, per-lane online logsumexp
//   k_rowstats: reduce 25 chunk partials -> per-row m, log(S)
//   k_out     : recompute tiles with WMMA, write logit - m - logS (~2.2us)
//
// K-order remap: WMMA step kk covers global K {2kk,2kk+1} via lanes 0-15
// (local k 0,1) and {32+2kk,32+2kk+1} via lanes 16-31 (local k 2,3), so each
// lane's A/B fragment is a CONTIGUOUS 32-float half-row -> 8 x b128 loads.
// ---------------------------------------------------------------------------

typedef float v2f __attribute__((ext_vector_type(2)));
typedef float v8f __attribute__((ext_vector_type(8)));
typedef float f4  __attribute__((ext_vector_type(4)));

#if __has_builtin(__builtin_amdgcn_sched_barrier)
#define SCHED_FENCE() __builtin_amdgcn_sched_barrier(0)
#else
#define SCHED_FENCE()
#endif

#define VV      50000
#define DD      64
#define BB      256
#define TWO_N   8
#define VCHUNKS 25          // 3125 tiles / 125 per chunk
#define TPC     125         // tiles per chunk
#define BBLOCKS 16

// ---------------------------------------------------------------- k_extract
__global__ __launch_bounds__(256) void k_extract(const f4* __restrict__ x4,
                                                 int* __restrict__ ids,
                                                 float* __restrict__ vals) {
  const long long n4      = (long long)TWO_N * BB * VV / 4;   // 25,600,000
  const long long tid     = (long long)blockIdx.x * blockDim.x + threadIdx.x;
  const long long stride  = (long long)gridDim.x * blockDim.x; // 320,000
  const long long stride4 = stride * 4;

  for (long long base = tid; base < n4; base += stride4) {
    f4 v0 = __builtin_nontemporal_load(&x4[base + 0 * stride]);
    f4 v1 = __builtin_nontemporal_load(&x4[base + 1 * stride]);
    f4 v2 = __builtin_nontemporal_load(&x4[base + 2 * stride]);
    f4 v3 = __builtin_nontemporal_load(&x4[base + 3 * stride]);
    f4 vv[4] = {v0, v1, v2, v3};
    #pragma unroll
    for (int u = 0; u < 4; ++u) {
      f4 v = vv[u];
      if (v.x != 0.f || v.y != 0.f || v.z != 0.f || v.w != 0.f) {
        float c[4] = {v.x, v.y, v.z, v.w};
        long long e0 = (base + u * stride) * 4;
        #pragma unroll
        for (int k = 0; k < 4; ++k) {
          if (c[k] != 0.f) {
            long long e = e0 + k;
            int ib   = (int)(e / VV);   // i*B + b (exactly one hot per ib)
            int vidx = (int)(e % VV);
            ids[ib]  = vidx;
            vals[ib] = c[k];
          }
        }
      }
    }
  }
}

// ----------------------------------------------------------------- k_hidden
__global__ __launch_bounds__(256) void k_hidden(const float* __restrict__ w1,
                                                const float* __restrict__ b1,
                                                const int* __restrict__ ids,
                                                const float* __restrict__ vals,
                                                float* __restrict__ h) {
  int t = blockIdx.x * blockDim.x + threadIdx.x;   // 16384 threads
  int b = t >> 6, d = t & 63;
  float acc = 0.f;
  #pragma unroll
  for (int i = 0; i < TWO_N; ++i) {
    int id = ids[i * BB + b];
    acc += vals[i * BB + b] * w1[(long long)d * VV + id];
  }
  h[t] = b1[d] + acc * 0.125f;
}

// Load a 32-float contiguous half-row as 8 x b128 into 16 v2f k-step frags.
__device__ __forceinline__ void load_frag32(v2f fr[16], const float* __restrict__ p) {
  const f4* p4 = (const f4*)p;
  #pragma unroll
  for (int i = 0; i < 8; ++i) {
    f4 q = p4[i];
    fr[2 * i + 0].x = q.x; fr[2 * i + 0].y = q.y;
    fr[2 * i + 1].x = q.z; fr[2 * i + 1].y = q.w;
  }
}

// 16 chained V_WMMA_F32_16X16X4_F32 on pre-loaded fragments.
__device__ __forceinline__ v8f tile_gemm(const v2f a[16], const v2f bf[16],
                                         float bias) {
  v8f acc;
  #pragma unroll
  for (int j = 0; j < 8; ++j) acc[j] = bias;
  #pragma unroll
  for (int kk = 0; kk < 16; ++kk) {
    acc = __builtin_amdgcn_wmma_f32_16x16x4_f32(false, a[kk], false, bf[kk],
                                                (short)0, acc, false, false);
  }
  return acc;
}

// ------------------------------------------------------------------ k_stats
__global__ __launch_bounds__(256) void k_stats(const float* __restrict__ h,
                                               const float* __restrict__ w2,
                                               const float* __restrict__ b2,
                                               float* __restrict__ pm,
                                               float* __restrict__ ps) {
  const int chunk = blockIdx.x;            // 0..24
  const int bblk  = blockIdx.y;            // 0..15
  const int tid   = threadIdx.x;
  const int wave  = tid >> 5;
  const int lane  = tid & 31;
  const int lh    = lane & 15;
  const int hi    = lane >> 4;
  const int b0    = bblk * 16;

  v2f a[16];                               // A reused across whole v loop
  load_frag32(a, h + (b0 + lh) * DD + hi * 32);

  float run_m[8], run_s[8];
  #pragma unroll
  for (int j = 0; j < 8; ++j) { run_m[j] = -__builtin_inff(); run_s[j] = 0.f; }

  const int tb = chunk * TPC;
  const int te = tb + TPC;

  auto fetch = [&](v2f* dst, float& bias, int tt) {
    int col = tt * 16 + lh;
    load_frag32(dst, w2 + (long long)col * DD + hi * 32);
    bias = b2[col];
  };
  auto process = [&](const v2f* bfp, float bias) {
    v8f acc = tile_gemm(a, bfp, bias);
    #pragma unroll
    for (int j = 0; j < 8; ++j) {
      float v  = acc[j];
      float nm = fmaxf(run_m[j], v);
      run_s[j] = run_s[j] * __expf(run_m[j] - nm) + __expf(v - nm);
      run_m[j] = nm;
    }
  };

  // Double-buffered, manually 2x-unrolled tile loop; clamped prefetch index
  // (redundant L2 reload on last iter) keeps EXEC uniform and all-ones.
  v2f bfA[16], bfB[16];
  float biasA, biasB;
  int t = tb + wave;
  fetch(bfA, biasA, t);
  for (;;) {
    int tn = (t + 8 < te) ? t + 8 : t;
    fetch(bfB, biasB, tn);
    SCHED_FENCE();                         // keep prefetch above the WMMAs
    process(bfA, biasA);
    if (t + 8 >= te) break;
    t = tn;

    tn = (t + 8 < te) ? t + 8 : t;
    fetch(bfA, biasA, tn);
    SCHED_FENCE();
    process(bfB, biasB);
    if (t + 8 >= te) break;
    t = tn;
  }

  // One cross-lane (m,s) merge per half-wave: masks 8,4,2,1 stay in-half.
  #pragma unroll
  for (int j = 0; j < 8; ++j) {
    float m = run_m[j], s = run_s[j];
    #pragma unroll
    for (int r = 0; r < 4; ++r) {
      int mask = 8 >> r;
      float om = __shfl_xor(m, mask);
      float os = __shfl_xor(s, mask);
      float nm = fmaxf(m, om);
      s = s * __expf(m - nm) + os * __expf(om - nm);
      m = nm;
    }
    run_m[j] = m; run_s[j] = s;
  }

  __shared__ float sm_m[8][16];
  __shared__ float sm_s[8][16];
  if (lh == 0) {
    #pragma unroll
    for (int j = 0; j < 8; ++j) {
      sm_m[wave][hi * 8 + j] = run_m[j];
      sm_s[wave][hi * 8 + j] = run_s[j];
    }
  }
  __syncthreads();
  if (tid < 16) {
    float m = -__builtin_inff();
    #pragma unroll
    for (int w = 0; w < 8; ++w) m = fmaxf(m, sm_m[w][tid]);
    float s = 0.f;
    #pragma unroll
    for (int w = 0; w < 8; ++w) s += sm_s[w][tid] * __expf(sm_m[w][tid] - m);
    int row = b0 + tid;
    pm[row * VCHUNKS + chunk] = m;
    ps[row * VCHUNKS + chunk] = s;
  }
}

// --------------------------------------------------------------- k_rowstats
__global__ __launch_bounds__(256) void k_rowstats(const float* __restrict__ pm,
                                                  const float* __restrict__ ps,
                                                  float* __restrict__ rm,
                                                  float* __restrict__ rls) {
  int b = blockIdx.x * blockDim.x + threadIdx.x;
  if (b >= BB) return;
  float m = -__builtin_inff();
  for (int c = 0; c < VCHUNKS; ++c) m = fmaxf(m, pm[b * VCHUNKS + c]);
  float s = 0.f;
  for (int c = 0; c < VCHUNKS; ++c)
    s += ps[b * VCHUNKS + c] * __expf(pm[b * VCHUNKS + c] - m);
  rm[b]  = m;
  rls[b] = logf(s);
}

// -------------------------------------------------------------------- k_out
__global__ __launch_bounds__(256) void k_out(const float* __restrict__ h,
                                             const float* __restrict__ w2,
                                             const float* __restrict__ b2,
                                             const float* __restrict__ rm,
                                             const float* __restrict__ rls,
                                             float* __restrict__ out) {
  const int chunk = blockIdx.x;
  const int bblk  = blockIdx.y;
  const int tid   = threadIdx.x;
  const int wave  = tid >> 5;
  const int lane  = tid & 31;
  const int lh    = lane & 15;
  const int hi    = lane >> 4;
  const int b0    = bblk * 16;

  v2f a[16];
  load_frag32(a, h + (b0 + lh) * DD + hi * 32);

  float sub[8];
  #pragma unroll
  for (int j = 0; j < 8; ++j) {
    int row = b0 + hi * 8 + j;
    sub[j] = rm[row] + rls[row];
  }

  const int tb = chunk * TPC;
  const int te = tb + TPC;

  auto fetch = [&](v2f* dst, float& bias, int tt) {
    int col = tt * 16 + lh;
    load_frag32(dst, w2 + (long long)col * DD + hi * 32);
    bias = b2[col];
  };
  auto process = [&](const v2f* bfp, float bias, int tt) {
    int col = tt * 16 + lh;
    v8f acc = tile_gemm(a, bfp, bias);
    #pragma unroll
    for (int j = 0; j < 8; ++j) {
      int row = b0 + hi * 8 + j;
      __builtin_nontemporal_store(acc[j] - sub[j],
                                  out + (long long)row * VV + col);
    }
  };

  v2f bfA[16], bfB[16];
  float biasA, biasB;
  int t = tb + wave;
  fetch(bfA, biasA, t);
  for (;;) {
    int tn = (t + 8 < te) ? t + 8 : t;
    fetch(bfB, biasB, tn);
    SCHED_FENCE();
    process(bfA, biasA, t);
    if (t + 8 >= te) break;
    t = tn;

    tn = (t + 8 < te) ? t + 8 : t;
    fetch(bfA, biasA, tn);
    SCHED_FENCE();
    process(bfB, biasB, t);
    if (t + 8 >= te) break;
    t = tn;
  }
}

// ---------------------------------------------------------------------------
extern "C" void kernel_launch(void* const* d_in, const int* in_sizes, int n_in,
                              void* d_out, int out_size, void* d_ws, size_t ws_size,
                              hipStream_t stream) {
  const float* x  = (const float*)d_in[0];   // [8,256,50000]
  const float* w1 = (const float*)d_in[1];   // [64,50000]
  const float* b1 = (const float*)d_in[2];   // [64]
  const float* w2 = (const float*)d_in[3];   // [50000,64]
  const float* b2 = (const float*)d_in[4];   // [50000]
  float* out = (float*)d_out;                // [256,50000]

  char* ws = (char*)d_ws;                    // ~135 KB total
  int*   ids  = (int*)  (ws + 0);            // 2048 ints
  float* vals = (float*)(ws + 8192);         // 2048 floats
  float* h    = (float*)(ws + 16384);        // 256*64
  float* pm   = (float*)(ws + 81920);        // 256*25
  float* ps   = (float*)(ws + 107520);       // 256*25
  float* rm   = (float*)(ws + 133120);       // 256
  float* rls  = (float*)(ws + 134144);       // 256

  k_extract<<<1250, 256, 0, stream>>>((const f4*)x, ids, vals);
  k_hidden <<<64,   256, 0, stream>>>(w1, b1, ids, vals, h);

  dim3 grid(VCHUNKS, BBLOCKS);
  k_stats   <<<grid, 256, 0, stream>>>(h, w2, b2, pm, ps);
  k_rowstats<<<1,    256, 0, stream>>>(pm, ps, rm, rls);
  k_out     <<<grid, 256, 0, stream>>>(h, w2, b2, rm, rls, out);
}